// DGCNN_cls_45406394253756
// MI455X (gfx1250) — compile-verified
//
#include <hip/hip_runtime.h>
#include <math.h>

typedef __attribute__((ext_vector_type(2))) float v2f;
typedef __attribute__((ext_vector_type(8))) float v8f;

#define NPTS 1024
#define KNN  20
#define BATCH 8
#define WPB 4                 // waves per block in WMMA kernels
#define BN_RSQ 0.99999500003749968f   // 1/sqrt(1 + 1e-5)

__device__ __forceinline__ float leaky(float v) { return v > 0.0f ? v : 0.2f * v; }

// Branchless guarded row load: unconditional load from clamped row, value
// selected to 0 when out of range (v_cndmask, no EXEC divergence).
template<bool GUARD>
__device__ __forceinline__ float rowld(const float* __restrict__ base, int k,
                                       int K, int col) {
    if (GUARD) {
        int kc = k < K ? k : K - 1;
        float v = base[(long)kc * NPTS + col];
        return k < K ? v : 0.0f;
    }
    return base[(long)k * NPTS + col];
}

// ---------------------------------------------------------------------------
// xx[b,n] = sum_c x[b,c,n]^2
// ---------------------------------------------------------------------------
__global__ void xx_kernel(const float* __restrict__ x, long xbs, int C,
                          float* __restrict__ xx) {
    int i = blockIdx.x * blockDim.x + threadIdx.x;          // b*NPTS + n
    if (i >= BATCH * NPTS) return;
    int b = i / NPTS, n = i % NPTS;
    const float* xb = x + (long)b * xbs + n;
    float s = 0.0f;
    for (int c = 0; c < C; ++c) { float v = xb[(long)c * NPTS]; s += v * v; }
    xx[i] = s;
}

// ---------------------------------------------------------------------------
// d[b,n,m] = 2*sum_c x[c,n]x[c,m] - xx[n] - xx[m]   via fp32 WMMA 16x16x4.
// One wave computes a 16x64 strip: one A fragment feeds 4 WMMAs per K-step.
// ---------------------------------------------------------------------------
template<bool GUARD>
__global__ void __launch_bounds__(WPB * 32)
gram_dist_kernel(const float* __restrict__ x, long xbs,
                 const float* __restrict__ xx,
                 float* __restrict__ d, int C) {
    const int tm = NPTS / 16, tn4 = NPTS / 64;
    const int tilesPerB = tm * tn4;
    int wid = blockIdx.x * WPB + (threadIdx.x >> 5);
    if (wid >= BATCH * tilesPerB) return;
    int b = wid / tilesPerB;
    int t = wid % tilesPerB;
    int m0 = (t / tn4) << 4;         // output row tile (n index)
    int n0 = (t % tn4) << 6;         // output col strip (m index), 64 wide
    int lane = threadIdx.x & 31;
    int half = lane >> 4;            // 0: K={0,1}, 1: K={2,3}
    int l16  = lane & 15;
    const float* xb = x + (long)b * xbs;

    v8f acc[4];
#pragma unroll
    for (int u = 0; u < 4; ++u) acc[u] = (v8f){0.f,0.f,0.f,0.f,0.f,0.f,0.f,0.f};

    for (int k = 0; k < C; k += 4) {
        int ka = k + half * 2;
        v2f a;
        a.x = rowld<GUARD>(xb, ka,     C, m0 + l16);
        a.y = rowld<GUARD>(xb, ka + 1, C, m0 + l16);
#pragma unroll
        for (int u = 0; u < 4; ++u) {
            v2f bb;
            bb.x = rowld<GUARD>(xb, ka,     C, n0 + (u << 4) + l16);
            bb.y = rowld<GUARD>(xb, ka + 1, C, n0 + (u << 4) + l16);
            acc[u] = __builtin_amdgcn_wmma_f32_16x16x4_f32(false, a, false, bb,
                                                           (short)0, acc[u],
                                                           false, false);
        }
    }
    const float* xxb = xx + b * NPTS;
    float* db = d + (long)b * NPTS * NPTS;
#pragma unroll
    for (int u = 0; u < 4; ++u) {
#pragma unroll
        for (int j = 0; j < 8; ++j) {
            int row = m0 + j + half * 8;
            int col = n0 + (u << 4) + l16;
            db[(long)row * NPTS + col] = 2.0f * acc[u][j] - xxb[row] - xxb[col];
        }
    }
}

// ---------------------------------------------------------------------------
// Generic fp32 WMMA GEMM: Y[b,m,n] = sum_k W[m, coff+k] * X[b,k,n]
// 16x64 strip per wave; optional fused BN + leaky epilogue.
// ---------------------------------------------------------------------------
template<bool GUARD>
__global__ void __launch_bounds__(WPB * 32)
wmma_gemm_kernel(const float* __restrict__ W, int lda, int coff,
                 const float* __restrict__ X, long xbs,
                 float* __restrict__ Y, long ybs,
                 int M, int K,
                 const float* __restrict__ bng,
                 const float* __restrict__ bnb, int doLeaky) {
    const int tn4 = NPTS / 64;
    int tm = M >> 4;
    int tilesPerB = tm * tn4;
    int wid = blockIdx.x * WPB + (threadIdx.x >> 5);
    if (wid >= BATCH * tilesPerB) return;
    int b = wid / tilesPerB;
    int t = wid % tilesPerB;
    int m0 = (t / tn4) << 4;
    int n0 = (t % tn4) << 6;
    int lane = threadIdx.x & 31;
    int half = lane >> 4;
    int l16  = lane & 15;
    const float* xb = X + (long)b * xbs;
    const float* wrow = W + (long)(m0 + l16) * lda + coff;

    v8f acc[4];
#pragma unroll
    for (int u = 0; u < 4; ++u) acc[u] = (v8f){0.f,0.f,0.f,0.f,0.f,0.f,0.f,0.f};

    for (int k = 0; k < K; k += 4) {
        int ka = k + half * 2;
        v2f a;
        if (GUARD) {
            float ax = wrow[ka     < K ? ka     : K - 1];
            float ay = wrow[ka + 1 < K ? ka + 1 : K - 1];
            a.x = ka     < K ? ax : 0.0f;
            a.y = ka + 1 < K ? ay : 0.0f;
        } else {
            a.x = wrow[ka];
            a.y = wrow[ka + 1];
        }
#pragma unroll
        for (int u = 0; u < 4; ++u) {
            v2f bb;
            bb.x = rowld<GUARD>(xb, ka,     K, n0 + (u << 4) + l16);
            bb.y = rowld<GUARD>(xb, ka + 1, K, n0 + (u << 4) + l16);
            acc[u] = __builtin_amdgcn_wmma_f32_16x16x4_f32(false, a, false, bb,
                                                           (short)0, acc[u],
                                                           false, false);
        }
    }
    float* yb = Y + (long)b * ybs;
#pragma unroll
    for (int u = 0; u < 4; ++u) {
#pragma unroll
        for (int j = 0; j < 8; ++j) {
            int row = m0 + j + half * 8;
            int col = n0 + (u << 4) + l16;
            float v = acc[u][j];
            if (bng) v = v * (bng[row] * BN_RSQ) + bnb[row];
            if (doLeaky) v = leaky(v);
            yb[(long)row * NPTS + col] = v;
        }
    }
}

// ---------------------------------------------------------------------------
// top-k(20) per row of d (descending, lowest-index ties) -> idx[b,n,0..19]
// One wave32 per row; 32 values per lane held in registers.
// ---------------------------------------------------------------------------
__global__ void topk_kernel(const float* __restrict__ d, int* __restrict__ idx) {
    int wid = blockIdx.x * (blockDim.x >> 5) + (threadIdx.x >> 5);
    if (wid >= BATCH * NPTS) return;
    int lane = threadIdx.x & 31;
    const float* row = d + (long)wid * NPTS;
    float v[32];
#pragma unroll
    for (int i = 0; i < 32; ++i) v[i] = row[lane + (i << 5)];
    int* orow = idx + (long)wid * KNN;

    for (int k = 0; k < KNN; ++k) {
        float bv = v[0]; int bi = lane;
#pragma unroll
        for (int i = 1; i < 32; ++i) {
            int col = lane + (i << 5);
            if (v[i] > bv) { bv = v[i]; bi = col; }      // strict > keeps lowest idx
        }
#pragma unroll
        for (int off = 16; off > 0; off >>= 1) {
            float ov = __shfl_xor(bv, off, 32);
            int   oi = __shfl_xor(bi, off, 32);
            if (ov > bv || (ov == bv && oi < bi)) { bv = ov; bi = oi; }
        }
        if (lane == 0) orow[k] = bi;
        int wi = bi >> 5, wl = bi & 31;
        if (wl == lane) {
#pragma unroll
            for (int i = 0; i < 32; ++i) if (i == wi) v[i] = -INFINITY;
        }
    }
}

// ---------------------------------------------------------------------------
// s[b,o,j,n] for j=0..2: leaky(bn(max over topk prefix {20,10,5} of
//   y1[o, idx[n,k]] + (yc-y1)[o,n]))
// ---------------------------------------------------------------------------
__global__ void smax_kernel(const float* __restrict__ y1,
                            const float* __restrict__ yc,
                            const int* __restrict__ idx,
                            const float* __restrict__ g,
                            const float* __restrict__ bb,
                            float* __restrict__ s, int O) {
    long i = (long)blockIdx.x * blockDim.x + threadIdx.x;   // ((b*O+o)*NPTS+n)
    if (i >= (long)BATCH * O * NPTS) return;
    int n = i % NPTS;
    long bo = i / NPTS;
    int o = bo % O, b = bo / O;
    const int* id = idx + ((long)b * NPTS + n) * KNN;
    const float* y1b = y1 + bo * NPTS;
    float base = yc[bo * NPTS + n] - y1b[n];
    float mx = -INFINITY, m5 = 0.0f, m10 = 0.0f;
#pragma unroll
    for (int k = 0; k < KNN; ++k) {
        mx = fmaxf(mx, y1b[id[k]]);
        if (k == 4) m5 = mx;
        if (k == 9) m10 = mx;
    }
    float scale = g[o] * BN_RSQ, bias = bb[o];
    float* so = s + bo * 3 * NPTS + n;
    so[0 * NPTS] = leaky((mx  + base) * scale + bias);   // j=0: max over 20
    so[1 * NPTS] = leaky((m10 + base) * scale + bias);   // j=1: max over 10
    so[2 * NPTS] = leaky((m5  + base) * scale + bias);   // j=2: max over 5
}

// ---------------------------------------------------------------------------
// Channel-attention pooling: mean & max of s over (j,n) per (b,o). Wave/row.
// ---------------------------------------------------------------------------
__global__ void capool_kernel(const float* __restrict__ s,
                              float* __restrict__ mavg, float* __restrict__ mmax,
                              int O) {
    int wid = blockIdx.x * (blockDim.x >> 5) + (threadIdx.x >> 5);
    if (wid >= BATCH * O) return;
    int lane = threadIdx.x & 31;
    const float* sb = s + (long)wid * 3 * NPTS;
    float sum = 0.0f, mx = -INFINITY;
    for (int i = lane; i < 3 * NPTS; i += 32) {
        float v = sb[i]; sum += v; mx = fmaxf(mx, v);
    }
#pragma unroll
    for (int off = 16; off > 0; off >>= 1) {
        sum += __shfl_xor(sum, off, 32);
        mx = fmaxf(mx, __shfl_xor(mx, off, 32));
    }
    if (lane == 0) { mavg[wid] = sum / (3.0f * NPTS); mmax[wid] = mx; }
}

// ---------------------------------------------------------------------------
// Channel-attention MLP: ca = sigmoid(w2@leaky(w1@avg) + w2@leaky(w1@max))
// One block per batch, O threads.
// ---------------------------------------------------------------------------
__global__ void camlp_kernel(const float* __restrict__ w1, const float* __restrict__ w2,
                             const float* __restrict__ mavg, const float* __restrict__ mmax,
                             float* __restrict__ ca, int O, int R) {
    int b = blockIdx.x, o = threadIdx.x;
    __shared__ float ha[16], hm[16];
    if (o < R) {
        float sa = 0.0f, sm = 0.0f;
        for (int c = 0; c < O; ++c) {
            float w = w1[o * O + c];
            sa += w * mavg[b * O + c];
            sm += w * mmax[b * O + c];
        }
        ha[o] = leaky(sa); hm[o] = leaky(sm);
    }
    __syncthreads();
    float va = 0.0f, vm = 0.0f;
    for (int r = 0; r < R; ++r) {
        float w = w2[o * R + r];
        va += w * ha[r]; vm += w * hm[r];
    }
    float t = va + vm;
    ca[b * O + o] = 1.0f / (1.0f + expf(-t));
}

// ---------------------------------------------------------------------------
// Spatial attention: sa[b,j,n] = sigmoid(w0*mean_o(ca*s) + w1*max_o(ca*s))
// ---------------------------------------------------------------------------
__global__ void sa_kernel(const float* __restrict__ s, const float* __restrict__ ca,
                          const float* __restrict__ saw, float* __restrict__ sa, int O) {
    int i = blockIdx.x * blockDim.x + threadIdx.x;    // (b*3+j)*NPTS + n
    if (i >= BATCH * 3 * NPTS) return;
    int n = i % NPTS;
    int bj = i / NPTS;
    int j = bj % 3, b = bj / 3;
    float sum = 0.0f, mx = -INFINITY;
    for (int o = 0; o < O; ++o) {
        float v = ca[b * O + o] * s[(((long)(b * O + o)) * 3 + j) * NPTS + n];
        sum += v; mx = fmaxf(mx, v);
    }
    float t = saw[0] * (sum / (float)O) + saw[1] * mx;
    sa[i] = 1.0f / (1.0f + expf(-t));
}

// ---------------------------------------------------------------------------
// x_out[b,o,n] = mean_j( sa[b,j,n] * ca[b,o] * s[b,o,j,n] )  -> hcat slice
// ---------------------------------------------------------------------------
__global__ void finalize_kernel(const float* __restrict__ s, const float* __restrict__ ca,
                                const float* __restrict__ sa, float* __restrict__ hcat,
                                int choff, int O) {
    long i = (long)blockIdx.x * blockDim.x + threadIdx.x;   // (b*O+o)*NPTS + n
    if (i >= (long)BATCH * O * NPTS) return;
    int n = i % NPTS;
    long bo = i / NPTS;
    int o = bo % O, b = bo / O;
    float c = ca[b * O + o];
    float acc = 0.0f;
#pragma unroll
    for (int j = 0; j < 3; ++j)
        acc += sa[(b * 3 + j) * NPTS + n] * s[(bo * 3 + j) * NPTS + n];
    hcat[((long)b * 512 + choff + o) * NPTS + n] = c * acc * (1.0f / 3.0f);
}

// ---------------------------------------------------------------------------
// Global pooling of conv5 output: feat[b, o]=max_n, feat[b, 1024+o]=mean_n
// ---------------------------------------------------------------------------
__global__ void pool_kernel(const float* __restrict__ h5, float* __restrict__ feat) {
    int wid = blockIdx.x * (blockDim.x >> 5) + (threadIdx.x >> 5);
    if (wid >= BATCH * 1024) return;
    int lane = threadIdx.x & 31;
    int b = wid / 1024, o = wid % 1024;
    const float* row = h5 + (long)wid * NPTS;
    float sum = 0.0f, mx = -INFINITY;
    for (int i = lane; i < NPTS; i += 32) {
        float v = row[i]; sum += v; mx = fmaxf(mx, v);
    }
#pragma unroll
    for (int off = 16; off > 0; off >>= 1) {
        sum += __shfl_xor(sum, off, 32);
        mx = fmaxf(mx, __shfl_xor(mx, off, 32));
    }
    if (lane == 0) {
        feat[b * 2048 + o]        = mx;
        feat[b * 2048 + 1024 + o] = sum / (float)NPTS;
    }
}

// ---------------------------------------------------------------------------
// Small FC: out[b,j] = act(bn(in[b,:]·w[j,:] + bias[j]))
// ---------------------------------------------------------------------------
__global__ void fc_kernel(const float* __restrict__ in, const float* __restrict__ w,
                          const float* __restrict__ bias,
                          const float* __restrict__ g, const float* __restrict__ bb,
                          int doLeaky, float* __restrict__ out, int IN, int OUT) {
    int i = blockIdx.x * blockDim.x + threadIdx.x;   // b*OUT + j
    if (i >= BATCH * OUT) return;
    int j = i % OUT, b = i / OUT;
    float s = 0.0f;
    for (int c = 0; c < IN; ++c) s += in[b * IN + c] * w[j * IN + c];
    if (bias) s += bias[j];
    if (g) s = s * (g[j] * BN_RSQ) + bb[j];
    if (doLeaky) s = leaky(s);
    out[i] = s;
}

// ---------------------------------------------------------------------------
extern "C" void kernel_launch(void* const* d_in, const int* in_sizes, int n_in,
                              void* d_out, int out_size, void* d_ws, size_t ws_size,
                              hipStream_t stream) {
    (void)in_sizes; (void)n_in; (void)out_size; (void)ws_size;
    const float* x       = (const float*)d_in[0];
    const float* convw[5]= {(const float*)d_in[1], (const float*)d_in[2],
                            (const float*)d_in[3], (const float*)d_in[4],
                            (const float*)d_in[5]};
    const float* caw1[4] = {(const float*)d_in[6],  (const float*)d_in[8],
                            (const float*)d_in[10], (const float*)d_in[12]};
    const float* caw2[4] = {(const float*)d_in[7],  (const float*)d_in[9],
                            (const float*)d_in[11], (const float*)d_in[13]};
    const float* saw[4]  = {(const float*)d_in[14], (const float*)d_in[15],
                            (const float*)d_in[16], (const float*)d_in[17]};
    const float* lin1_w = (const float*)d_in[18];
    const float* lin2_w = (const float*)d_in[19];
    const float* lin2_b = (const float*)d_in[20];
    const float* lin3_w = (const float*)d_in[21];
    const float* lin3_b = (const float*)d_in[22];
    const float* bng[7], *bnb[7];
    for (int i = 0; i < 7; ++i) {
        bng[i] = (const float*)d_in[23 + 2 * i];
        bnb[i] = (const float*)d_in[24 + 2 * i];
    }

    // ---- workspace layout (~93 MB; entire pipeline fits in the 192 MB L2) ----
    char* p = (char*)d_ws;
    auto alloc = [&](size_t bytes) -> void* {
        void* r = (void*)p; p += (bytes + 255) & ~(size_t)255; return r;
    };
    float* dmat = (float*)alloc((size_t)BATCH * NPTS * NPTS * 4);   // dist / conv5 out (reused)
    float* xxb  = (float*)alloc((size_t)BATCH * NPTS * 4);
    int*   idxb = (int*)  alloc((size_t)BATCH * NPTS * KNN * 4);
    float* y1   = (float*)alloc((size_t)BATCH * 256 * NPTS * 4);
    float* yc   = (float*)alloc((size_t)BATCH * 256 * NPTS * 4);
    float* sbuf = (float*)alloc((size_t)BATCH * 256 * 3 * NPTS * 4);
    float* mavg = (float*)alloc((size_t)BATCH * 256 * 4);
    float* mmax = (float*)alloc((size_t)BATCH * 256 * 4);
    float* cab  = (float*)alloc((size_t)BATCH * 256 * 4);
    float* sab  = (float*)alloc((size_t)BATCH * 3 * NPTS * 4);
    float* hcat = (float*)alloc((size_t)BATCH * 512 * NPTS * 4);
    float* feat = (float*)alloc((size_t)BATCH * 2048 * 4);
    float* fca  = (float*)alloc((size_t)BATCH * 512 * 4);
    float* fcb  = (float*)alloc((size_t)BATCH * 256 * 4);

    auto edge_block = [&](const float* xin, long xbs, int Cin, int O,
                          const float* cw, const float* w1, const float* w2, int R,
                          const float* sw, const float* g, const float* bb,
                          int choff, bool guard) {
        xx_kernel<<<(BATCH * NPTS + 255) / 256, 256, 0, stream>>>(xin, xbs, Cin, xxb);

        int gw = BATCH * (NPTS / 16) * (NPTS / 64);
        if (guard)
            gram_dist_kernel<true><<<(gw + WPB - 1) / WPB, WPB * 32, 0, stream>>>(
                xin, xbs, xxb, dmat, Cin);
        else
            gram_dist_kernel<false><<<(gw + WPB - 1) / WPB, WPB * 32, 0, stream>>>(
                xin, xbs, xxb, dmat, Cin);

        topk_kernel<<<(BATCH * NPTS + 7) / 8, 256, 0, stream>>>(dmat, idxb);

        int cv = BATCH * (O / 16) * (NPTS / 64);
        if (guard) {
            wmma_gemm_kernel<true><<<(cv + WPB - 1) / WPB, WPB * 32, 0, stream>>>(
                cw, 2 * Cin, 0,   xin, xbs, y1, (long)O * NPTS, O, Cin, nullptr, nullptr, 0);
            wmma_gemm_kernel<true><<<(cv + WPB - 1) / WPB, WPB * 32, 0, stream>>>(
                cw, 2 * Cin, Cin, xin, xbs, yc, (long)O * NPTS, O, Cin, nullptr, nullptr, 0);
        } else {
            wmma_gemm_kernel<false><<<(cv + WPB - 1) / WPB, WPB * 32, 0, stream>>>(
                cw, 2 * Cin, 0,   xin, xbs, y1, (long)O * NPTS, O, Cin, nullptr, nullptr, 0);
            wmma_gemm_kernel<false><<<(cv + WPB - 1) / WPB, WPB * 32, 0, stream>>>(
                cw, 2 * Cin, Cin, xin, xbs, yc, (long)O * NPTS, O, Cin, nullptr, nullptr, 0);
        }

        long se = (long)BATCH * O * NPTS;
        smax_kernel<<<(se + 255) / 256, 256, 0, stream>>>(y1, yc, idxb, g, bb, sbuf, O);

        capool_kernel<<<(BATCH * O + 7) / 8, 256, 0, stream>>>(sbuf, mavg, mmax, O);
        camlp_kernel<<<BATCH, O, 0, stream>>>(w1, w2, mavg, mmax, cab, O, R);
        sa_kernel<<<(BATCH * 3 * NPTS + 255) / 256, 256, 0, stream>>>(sbuf, cab, sw, sab, O);
        finalize_kernel<<<(se + 255) / 256, 256, 0, stream>>>(sbuf, cab, sab, hcat, choff, O);
    };

    // Blocks 2-4 read their input directly as a channel slice of hcat.
    edge_block(x,                  3L * NPTS,     3,  64, convw[0], caw1[0], caw2[0],  4, saw[0], bng[0], bnb[0],   0, true);
    edge_block(hcat,               512L * NPTS,  64,  64, convw[1], caw1[1], caw2[1],  4, saw[1], bng[1], bnb[1],  64, false);
    edge_block(hcat + 64L * NPTS,  512L * NPTS,  64, 128, convw[2], caw1[2], caw2[2],  8, saw[2], bng[2], bnb[2], 128, false);
    edge_block(hcat + 128L * NPTS, 512L * NPTS, 128, 256, convw[3], caw1[3], caw2[3], 16, saw[3], bng[3], bnb[3], 256, false);

    // conv5 (1024x512 GEMM, fused bn5+leaky) -> reuse dmat as output buffer
    int c5 = BATCH * (1024 / 16) * (NPTS / 64);
    wmma_gemm_kernel<false><<<(c5 + WPB - 1) / WPB, WPB * 32, 0, stream>>>(
        convw[4], 512, 0, hcat, 512L * NPTS, dmat, 1024L * NPTS, 1024, 512,
        bng[4], bnb[4], 1);

    pool_kernel<<<(BATCH * 1024 + 7) / 8, 256, 0, stream>>>(dmat, feat);

    fc_kernel<<<(BATCH * 512 + 255) / 256, 256, 0, stream>>>(
        feat, lin1_w, nullptr, bng[5], bnb[5], 1, fca, 2048, 512);
    fc_kernel<<<(BATCH * 256 + 255) / 256, 256, 0, stream>>>(
        fca, lin2_w, lin2_b, bng[6], bnb[6], 1, fcb, 512, 256);
    fc_kernel<<<(BATCH * 40 + 255) / 256, 256, 0, stream>>>(
        fcb, lin3_w, lin3_b, nullptr, nullptr, 0, (float*)d_out, 256, 40);
}